// RGCN_82059645157462
// MI455X (gfx1250) — compile-verified
//
#include <hip/hip_runtime.h>

// ---------------------------------------------------------------------------
// RGCN (N=50000, E=600000, D=64, R=65, L=2) for gfx1250 (CDNA5, wave32).
// All matmuls run on V_WMMA_F32_16X16X4_F32 (full fp32 accuracy).
// Strategy:
//   1) counting-sort edges by relation (padded to 64-edge tiles)
//   2) per-tile dense [64x64]@[64x64] WMMA GEMM: msg = x[src] @ W[rel],
//      scaled by 1/cnt(dst,rel), atomic-added into conv accumulator
//   3) per-64-node-tile WMMA GEMM for h@root and h@res_W with fused
//      relu/bias epilogue -> next layer h
// ---------------------------------------------------------------------------

typedef __attribute__((ext_vector_type(2))) float v2f;
typedef __attribute__((ext_vector_type(8))) float v8f;

#define DD 64
#define AP 68   // LDS pitch for A tiles (4*M + c distinct banks across lanes)
#define WP 72   // LDS pitch for W tiles (half-wave rows land in disjoint banks)

__device__ __forceinline__ v8f wmma4(v2f a, v2f b, v8f c) {
  // D = A(16x4,f32) * B(4x16,f32) + C(16x16,f32)
  return __builtin_amdgcn_wmma_f32_16x16x4_f32(
      /*neg_a=*/false, a, /*neg_b=*/false, b,
      /*c_mod=*/(short)0, c, /*reuse_a=*/false, /*reuse_b=*/false);
}

// --- pass 1: per-(dst,rel) degree count + per-relation histogram -----------
__global__ void edge_prep(const int* __restrict__ dst, const int* __restrict__ et,
                          int* __restrict__ cnt, int* __restrict__ hist,
                          int E, int R) {
  int e = blockIdx.x * blockDim.x + threadIdx.x;
  if (e >= E) return;
  int t = et[e];
  atomicAdd(&cnt[(size_t)dst[e] * R + t], 1);
  atomicAdd(&hist[t], 1);
}

// --- pass 2: serial scan over R=65 buckets, pad each to 64-edge tiles ------
__global__ void scan_pad(const int* __restrict__ hist, int* __restrict__ cursor,
                         int* __restrict__ tileRel, int R, int maxTiles) {
  if (blockIdx.x != 0 || threadIdx.x != 0) return;
  int run = 0;  // running tile count
  for (int r = 0; r < R; ++r) {
    cursor[r] = run * 64;            // first slot for relation r
    int tiles = (hist[r] + 63) >> 6;
    for (int tt = 0; tt < tiles; ++tt) tileRel[run + tt] = r;
    run += tiles;
  }
  for (int tt = run; tt < maxTiles; ++tt) tileRel[tt] = -1;
}

// --- pass 3: scatter edge ids into relation-grouped, padded slots ----------
__global__ void edge_scatter(const int* __restrict__ et, int* __restrict__ cursor,
                             int* __restrict__ perm, int E) {
  int e = blockIdx.x * blockDim.x + threadIdx.x;
  if (e >= E) return;
  int pos = atomicAdd(&cursor[et[e]], 1);
  perm[pos] = e;
}

// --- per-tile edge transform: conv[dst] += (x[src] @ W[rel]) / cnt ---------
__global__ void __launch_bounds__(256)
edge_gemm(const float* __restrict__ h, const float* __restrict__ Wl,
          const int* __restrict__ perm, const int* __restrict__ tileRel,
          const int* __restrict__ src, const int* __restrict__ dstArr,
          const int* __restrict__ et, const int* __restrict__ cnt,
          float* __restrict__ conv, int R) {
  __shared__ float As[64][AP];
  __shared__ float Ws[64][WP];
  __shared__ int   eSrc[64];
  __shared__ int   eDst[64];
  __shared__ float eInv[64];

  int rel = tileRel[blockIdx.x];
  if (rel < 0) return;                       // uniform: whole block exits
  int tid = threadIdx.x;

  // edge metadata for this tile (padding slots hold -1)
  if (tid < 64) {
    int e = perm[(size_t)blockIdx.x * 64 + tid];
    int s = -1, d = 0; float inv = 0.f;
    if (e >= 0) {
      s = src[e];
      d = dstArr[e];
      int c = cnt[(size_t)d * R + et[e]];
      inv = 1.0f / (float)(c > 0 ? c : 1);
    }
    eSrc[tid] = s; eDst[tid] = d; eInv[tid] = inv;
  }

  // stage W[rel] (64x64 = 4096 floats): 256 threads x 4 float4
  {
    const float4* W4 = (const float4*)(Wl + (size_t)rel * DD * DD);
    for (int i = 0; i < 4; ++i) {
      int idx = tid * 4 + i;                 // float4 index, 16 per row
      int r = idx >> 4, c4 = (idx & 15) * 4;
      float4 v = W4[idx];
      Ws[r][c4 + 0] = v.x; Ws[r][c4 + 1] = v.y;
      Ws[r][c4 + 2] = v.z; Ws[r][c4 + 3] = v.w;
    }
  }
  __syncthreads();

  // stage A (gathered source features): thread -> row tid>>2, 16 floats
  {
    int row = tid >> 2, cb = (tid & 3) * 16;
    int s = eSrc[row];
    if (s >= 0) {
      const float4* hp = (const float4*)(h + (size_t)s * DD + cb);
      for (int i = 0; i < 4; ++i) {
        float4 v = hp[i];
        As[row][cb + i * 4 + 0] = v.x; As[row][cb + i * 4 + 1] = v.y;
        As[row][cb + i * 4 + 2] = v.z; As[row][cb + i * 4 + 3] = v.w;
      }
    } else {
      for (int i = 0; i < 16; ++i) As[row][cb + i] = 0.f;
    }
  }
  __syncthreads();

  // 8 waves: wave w -> rows [16*(w>>1), +16), cols {32*(w&1), 32*(w&1)+16}
  int lane = tid & 31, w = tid >> 5;
  int rt = w >> 1, cp = w & 1;
  int cb0 = cp * 32, cb1 = cb0 + 16;
  int n = lane & 15, hi = lane >> 4;
  int m = rt * 16 + n;

  v8f c0 = {}; v8f c1 = {};
  for (int k = 0; k < 16; ++k) {
    int kb = k * 4 + hi * 2;
    v2f a;  a.x  = As[m][kb];        a.y  = As[m][kb + 1];
    v2f b0; b0.x = Ws[kb][cb0 + n];  b0.y = Ws[kb + 1][cb0 + n];
    v2f b1; b1.x = Ws[kb][cb1 + n];  b1.y = Ws[kb + 1][cb1 + n];
    c0 = wmma4(a, b0, c0);
    c1 = wmma4(a, b1, c1);
  }

  // scatter: row gr of C -> edge gr of tile; scale by 1/cnt, atomic add
  for (int i = 0; i < 8; ++i) {
    int gr = rt * 16 + hi * 8 + i;
    if (eSrc[gr] >= 0) {
      float inv = eInv[gr];
      size_t base = (size_t)eDst[gr] * DD;
      atomicAdd(&conv[base + cb0 + n], c0[i] * inv);
      atomicAdd(&conv[base + cb1 + n], c1[i] * inv);
    }
  }
}

// --- dense node update: h' = relu(conv + h@root + b) + relu(h@resW + rb) ---
__global__ void __launch_bounds__(256)
node_gemm(const float* __restrict__ h, const float* __restrict__ conv,
          const float* __restrict__ rootW, const float* __restrict__ biasv,
          const float* __restrict__ resW, const float* __restrict__ resb,
          float* __restrict__ hout, int N) {
  __shared__ float As[64][AP];
  __shared__ float Wr[64][WP];
  __shared__ float Wq[64][WP];
  __shared__ float bV[64], rbV[64];

  int tid = threadIdx.x;
  int base = blockIdx.x * 64;

  if (tid < 64) { bV[tid] = biasv[tid]; rbV[tid] = resb[tid]; }

  for (int i = 0; i < 4; ++i) {
    int idx = tid * 4 + i;
    int r = idx >> 4, c4 = (idx & 15) * 4;
    float4 v = ((const float4*)rootW)[idx];
    Wr[r][c4 + 0] = v.x; Wr[r][c4 + 1] = v.y; Wr[r][c4 + 2] = v.z; Wr[r][c4 + 3] = v.w;
    float4 u = ((const float4*)resW)[idx];
    Wq[r][c4 + 0] = u.x; Wq[r][c4 + 1] = u.y; Wq[r][c4 + 2] = u.z; Wq[r][c4 + 3] = u.w;
  }

  {
    int row = tid >> 2, cb = (tid & 3) * 16;
    int node = base + row;
    if (node < N) {
      const float4* hp = (const float4*)(h + (size_t)node * DD + cb);
      for (int i = 0; i < 4; ++i) {
        float4 v = hp[i];
        As[row][cb + i * 4 + 0] = v.x; As[row][cb + i * 4 + 1] = v.y;
        As[row][cb + i * 4 + 2] = v.z; As[row][cb + i * 4 + 3] = v.w;
      }
    } else {
      for (int i = 0; i < 16; ++i) As[row][cb + i] = 0.f;
    }
  }
  __syncthreads();

  int lane = tid & 31, w = tid >> 5;
  int rt = w >> 1, cp = w & 1;
  int cb0 = cp * 32, cb1 = cb0 + 16;
  int n = lane & 15, hi = lane >> 4;
  int m = rt * 16 + n;

  v8f cR0 = {}, cR1 = {}, cS0 = {}, cS1 = {};
  for (int k = 0; k < 16; ++k) {
    int kb = k * 4 + hi * 2;
    v2f a;   a.x   = As[m][kb];        a.y   = As[m][kb + 1];
    v2f br0; br0.x = Wr[kb][cb0 + n];  br0.y = Wr[kb + 1][cb0 + n];
    v2f br1; br1.x = Wr[kb][cb1 + n];  br1.y = Wr[kb + 1][cb1 + n];
    v2f bq0; bq0.x = Wq[kb][cb0 + n];  bq0.y = Wq[kb + 1][cb0 + n];
    v2f bq1; bq1.x = Wq[kb][cb1 + n];  bq1.y = Wq[kb + 1][cb1 + n];
    cR0 = wmma4(a, br0, cR0);
    cR1 = wmma4(a, br1, cR1);
    cS0 = wmma4(a, bq0, cS0);
    cS1 = wmma4(a, bq1, cS1);
  }

  for (int i = 0; i < 8; ++i) {
    int gr = rt * 16 + hi * 8 + i;
    int node = base + gr;
    if (node < N) {
      size_t b0 = (size_t)node * DD + cb0 + n;
      size_t b1 = (size_t)node * DD + cb1 + n;
      float o0 = fmaxf(conv[b0] + cR0[i] + bV[cb0 + n], 0.f) +
                 fmaxf(cS0[i] + rbV[cb0 + n], 0.f);
      float o1 = fmaxf(conv[b1] + cR1[i] + bV[cb1 + n], 0.f) +
                 fmaxf(cS1[i] + rbV[cb1 + n], 0.f);
      hout[b0] = o0;
      hout[b1] = o1;
    }
  }
}

extern "C" void kernel_launch(void* const* d_in, const int* in_sizes, int n_in,
                              void* d_out, int out_size, void* d_ws, size_t ws_size,
                              hipStream_t stream) {
  const float* x    = (const float*)d_in[0];
  const int*   ei   = (const int*)d_in[1];   // [2,E]: row0=src, row1=dst
  const int*   et   = (const int*)d_in[2];   // [E]
  const float* W    = (const float*)d_in[3]; // [L,R,64,64]
  const float* root = (const float*)d_in[4]; // [L,64,64]
  const float* bias = (const float*)d_in[5]; // [L,64]
  const float* resW = (const float*)d_in[6]; // [L,64,64]
  const float* resb = (const float*)d_in[7]; // [L,64]

  const int N = in_sizes[0] / DD;
  const int E = in_sizes[2];
  const int L = in_sizes[4] / (DD * DD);
  const int R = in_sizes[3] / (L * DD * DD);
  const int* src  = ei;
  const int* dstA = ei + E;
  const int maxTiles = (E + 63) / 64 + R;

  // workspace carve-out (256B aligned chunks)
  char* wsb = (char*)d_ws;
  size_t off = 0;
  auto carve = [&](size_t bytes) -> void* {
    void* p = wsb + off;
    off += (bytes + 255) & ~(size_t)255;
    return p;
  };
  int*   cnt     = (int*)carve((size_t)N * R * sizeof(int));
  int*   hist    = (int*)carve((size_t)R * sizeof(int));
  int*   cursor  = (int*)carve((size_t)R * sizeof(int));
  int*   tileRel = (int*)carve((size_t)maxTiles * sizeof(int));
  int*   perm    = (int*)carve((size_t)maxTiles * 64 * sizeof(int));
  float* hA      = (float*)carve((size_t)N * DD * sizeof(float));
  float* hB      = (float*)carve((size_t)N * DD * sizeof(float));
  float* conv    = (float*)carve((size_t)N * DD * sizeof(float));

  hipMemsetAsync(cnt, 0, (size_t)N * R * sizeof(int), stream);
  hipMemsetAsync(hist, 0, (size_t)R * sizeof(int), stream);
  hipMemsetAsync(perm, 0xFF, (size_t)maxTiles * 64 * sizeof(int), stream); // -1

  const int eb = (E + 255) / 256;
  edge_prep   <<<eb, 256, 0, stream>>>(dstA, et, cnt, hist, E, R);
  scan_pad    <<<1, 32, 0, stream>>>(hist, cursor, tileRel, R, maxTiles);
  edge_scatter<<<eb, 256, 0, stream>>>(et, cursor, perm, E);

  const float* hcur = x;
  for (int l = 0; l < L; ++l) {
    float* hnext = (l == L - 1) ? (float*)d_out : ((l & 1) ? hB : hA);
    hipMemsetAsync(conv, 0, (size_t)N * DD * sizeof(float), stream);
    edge_gemm<<<maxTiles, 256, 0, stream>>>(
        hcur, W + (size_t)l * R * DD * DD, perm, tileRel,
        src, dstA, et, cnt, conv, R);
    node_gemm<<<(N + 63) / 64, 256, 0, stream>>>(
        hcur, conv, root + (size_t)l * DD * DD, bias + (size_t)l * DD,
        resW + (size_t)l * DD * DD, resb + (size_t)l * DD, hnext, N);
    hcur = hnext;
  }
}